// LSTM_49349174231230
// MI455X (gfx1250) — compile-verified
//
#include <hip/hip_runtime.h>
#include <math.h>

// Problem constants (reference: L=128, N=8192, H=20, C=3)
#define LSEQ  128
#define NBAT  8192
#define HID   20
#define GDIM  80          // 4*H
#define NCRV  3
#define LN3   (LSEQ * NBAT * NCRV)
#define TPB   8           // row-tiles per block in k_pre / k_post

typedef float v2f __attribute__((ext_vector_type(2)));
typedef float v8f __attribute__((ext_vector_type(8)));

// ---------------------------------------------------------------------------
// CDNA5 WMMA: D = A(16x4 f32) * B(4x16 f32) + C(16x16 f32), full fp32.
// H=20 = 5 K-chunks of 4, 4H=80 = 5 N-tiles of 16 -> exact fit, no padding.
// ---------------------------------------------------------------------------
__device__ __forceinline__ v8f wmma4(v2f a, v2f b, v8f c) {
  return __builtin_amdgcn_wmma_f32_16x16x4_f32(
      /*neg_a=*/false, a, /*neg_b=*/false, b,
      /*c_mod=*/(short)0, c, /*reuse_a=*/false, /*reuse_b=*/false);
}

// Fast transcendental forms (v_rcp_f32 / v_exp_f32, no IEEE div sequences):
// these sit on the serial 128-step recurrence, latency matters.
__device__ __forceinline__ float rcpf_(float x)     { return __builtin_amdgcn_rcpf(x); }
__device__ __forceinline__ float sigmoidf_(float x) { return rcpf_(1.0f + __expf(-x)); }
__device__ __forceinline__ float tanhf_(float x)    { return 1.0f - 2.0f * rcpf_(1.0f + __expf(2.0f * x)); }
__device__ __forceinline__ float siluf_(float x)    { return x * sigmoidf_(x); }

// 8-byte vector load (all call sites have even float index -> 8B aligned)
__device__ __forceinline__ v2f loadv2(const float* __restrict__ p) {
  return *(const v2f*)p;
}

// A-fragment (16 rows x K4 chunk) from an LDS tile, row stride `astride`.
// ISA layout: lanes 0-15 hold K=4kc,4kc+1 ; lanes 16-31 hold K=4kc+2,4kc+3.
__device__ __forceinline__ v2f load_afrag(const float* As, int astride, int kc, int lane) {
  int r  = lane & 15;
  int kb = 4 * kc + ((lane >> 4) << 1);
  v2f a;
  a.x = As[r * astride + kb];
  a.y = As[r * astride + kb + 1];
  return a;
}

// C/D fragment scatter to LDS (no guards: dest tiles are stride-padded so all
// 32 columns are writable; garbage columns are simply never read).
// v[k] = D[row k + 8*(lane>=16)][col = ot*16 + (lane&15)]
__device__ __forceinline__ void store_cfrag(float* Gs, int gstride, int ot,
                                            int lane, v8f c) {
  int j  = ot * 16 + (lane & 15);
  int rb = (lane >> 4) * 8;
#pragma unroll
  for (int k = 0; k < 8; k++) Gs[(rb + k) * gstride + j] = c[k];
}

// ---------------------------------------------------------------------------
// Pre-staged weight fragments (B = W^T, W row-major [OUT][Kw]) + bias, kept in
// VGPRs. Index-CLAMPED unconditional loads: no exec-mask branches. Clamped
// columns only produce discarded output columns; clamped K rows multiply
// against A's zero padding.
// ---------------------------------------------------------------------------
template <int NKC, int NOT>
struct BW {
  v2f  f[NKC][NOT];
  float b[NOT];
};

template <int NKC, int NOT>
__device__ __forceinline__ void load_bw(BW<NKC, NOT>& F, const float* __restrict__ W,
                                        int Kw, int OUT, const float* __restrict__ bias,
                                        int lane) {
  int kb  = ((lane >> 4) << 1);
#pragma unroll
  for (int ot = 0; ot < NOT; ot++) {
    int j  = ot * 16 + (lane & 15);
    int jc = j < OUT ? j : OUT - 1;            // clamp, no branch
    F.b[ot] = bias[jc];
#pragma unroll
    for (int kc = 0; kc < NKC; kc++) {
      int k  = 4 * kc + kb;
      int kcl = k <= Kw - 2 ? k : Kw - 2;      // clamp, no branch
      F.f[kc][ot] = loadv2(W + jc * Kw + kcl);
    }
  }
}

// Generic 16-row tile GEMM with pre-staged weights: Gs[16][*] = As @ W^T + b
template <int NKC, int NOT>
__device__ __forceinline__ void gemm_tile(const float* As, int astride,
                                          const BW<NKC, NOT>& F,
                                          float* Gs, int gstride, int lane) {
  v2f a[NKC];
#pragma unroll
  for (int kc = 0; kc < NKC; kc++) a[kc] = load_afrag(As, astride, kc, lane);
#pragma unroll
  for (int ot = 0; ot < NOT; ot++) {
    v8f c;
#pragma unroll
    for (int k = 0; k < 8; k++) c[k] = F.b[ot];
#pragma unroll
    for (int kc = 0; kc < NKC; kc++) c = wmma4(a[kc], F.f[kc][ot], c);
    store_cfrag(Gs, gstride, ot, lane, c);
  }
}

// Per-lane pointwise mapping over a 16x20 tile: lane owns (row = lane>>1,
// elements h0..h0+9 with h0 = (lane&1)*10).  32 lanes * 10 = 320 = 16*20.
__device__ __forceinline__ void silu_tile(float* T, int stride, int lane) {
  int row = lane >> 1, h0 = (lane & 1) * 10;
#pragma unroll
  for (int j = 0; j < 10; j++) {
    float x = T[row * stride + h0 + j];
    T[row * stride + h0 + j] = siluf_(x);
  }
}

// LayerNorm over H=20 per row; pair of lanes (2r,2r+1) cooperates via shfl_xor.
__device__ __forceinline__ void layernorm_tile(const float* In, int istride,
                                               float* Out, int ostride,
                                               const float* __restrict__ g,
                                               const float* __restrict__ b, int lane) {
  int row = lane >> 1, h0 = (lane & 1) * 10;
  float v[10];
  float s = 0.f, s2 = 0.f;
#pragma unroll
  for (int j = 0; j < 10; j++) {
    float x = In[row * istride + h0 + j];
    v[j] = x; s += x; s2 += x * x;
  }
  s  += __shfl_xor(s, 1, 32);
  s2 += __shfl_xor(s2, 1, 32);
  float mean = s * (1.0f / 20.0f);
  float var  = s2 * (1.0f / 20.0f) - mean * mean;
  float rstd = rsqrtf(var + 1e-5f);
#pragma unroll
  for (int j = 0; j < 10; j++) {
    float y = (v[j] - mean) * rstd * g[h0 + j];
    if (b) y += b[h0 + j];
    Out[row * ostride + h0 + j] = y;
  }
}

// LSTM cell pointwise: gates Gs[16][80] order i,f,g,o ; c in regs (10/lane);
// writes h to Hs and accumulates residual into Xs.
__device__ __forceinline__ void lstm_pointwise(const float* Gs, float* Hs, float* Xs,
                                               float* cst, int lane) {
  int row = lane >> 1, h0 = (lane & 1) * 10;
#pragma unroll
  for (int j = 0; j < 10; j++) {
    int h = h0 + j;
    float ig = Gs[row * GDIM + h];
    float fg = Gs[row * GDIM + 20 + h];
    float gg = Gs[row * GDIM + 40 + h];
    float og = Gs[row * GDIM + 60 + h];
    float cn = sigmoidf_(fg) * cst[j] + sigmoidf_(ig) * tanhf_(gg);
    cst[j] = cn;
    float hn = sigmoidf_(og) * tanhf_(cn);
    Hs[row * HID + h] = hn;
    Xs[row * HID + h] += hn;   // residual x = x + lstm(...)
  }
}

// Fused gate GEMM: Gs = bias + LN(x)@Wih^T + h@Whh^T  (weights pre-staged in VGPRs)
__device__ __forceinline__ void lstm_gates(const float* As, const float* Hs,
                                           const v2f (&Fx)[5][5], const v2f (&Fh)[5][5],
                                           const float (&gb)[5], float* Gs, int lane) {
  v2f ax[5], ah[5];
#pragma unroll
  for (int kc = 0; kc < 5; kc++) {
    ax[kc] = load_afrag(As, HID, kc, lane);
    ah[kc] = load_afrag(Hs, HID, kc, lane);
  }
#pragma unroll
  for (int ot = 0; ot < 5; ot++) {
    v8f c;
#pragma unroll
    for (int k = 0; k < 8; k++) c[k] = gb[ot];
#pragma unroll
    for (int kc = 0; kc < 5; kc++) c = wmma4(ax[kc], Fx[kc][ot], c);
#pragma unroll
    for (int kc = 0; kc < 5; kc++) c = wmma4(ah[kc], Fh[kc][ot], c);
    store_cfrag(Gs, GDIM, ot, lane, c);
  }
}

// ===========================================================================
// K1: preprocessing + lin1 + silu + ln1 + lin2 + silu  ->  xbuf [L,N,20]
// One wave per block, TPB row-tiles of 16; weight fragments hoisted into
// VGPRs across tiles. Single-wave workgroup: barriers are S_NOP-class fences.
// ===========================================================================
__global__ __launch_bounds__(32) void k_pre(
    const float* __restrict__ x_lni, const float* __restrict__ imean,
    const float* __restrict__ istd,
    const float* __restrict__ w1, const float* __restrict__ b1,
    const float* __restrict__ g1,
    const float* __restrict__ w2, const float* __restrict__ b2,
    float* __restrict__ xbuf) {
  __shared__ float Fin[16][8];
  __shared__ float T0[16][32];   // GEMM dest: stride 32 so stores are unguarded
  __shared__ float T1[16][20];
  int lane = threadIdx.x;
  int row = lane >> 1, h0 = (lane & 1) * 10;

  BW<2, 2> B1; load_bw(B1, w1,   6, HID, b1, lane);   // lin1 (K padded 6->8)
  BW<5, 2> B2; load_bw(B2, w2, HID, HID, b2, lane);   // lin2
  float m0 = imean[0], m1 = imean[1];
  float r0 = rcpf_(istd[0]), r1 = rcpf_(istd[1]);

  for (int tt = 0; tt < TPB; tt++) {
    long long row0 = ((long long)blockIdx.x * TPB + tt) * 16;
    __syncthreads();
    if (lane < 16) {
      long long r = row0 + lane;
      float delay  = x_lni[r * 3 + 0];
      float dur    = x_lni[r * 3 + 1];
      float rating = x_lni[r * 3 + 2];
      float f0 = (__logf(1e-5f + delay) - m0) * r0;
      float f1 = (__logf(fminf(fmaxf(dur, 100.0f), 60000.0f)) - m1) * r1;
      int ri = (int)fmaxf(rating, 1.0f) - 1;
      Fin[lane][0] = f0;
      Fin[lane][1] = f1;
#pragma unroll
      for (int k = 0; k < 4; k++) Fin[lane][2 + k] = (ri == k) ? 1.0f : 0.0f;
      Fin[lane][6] = 0.0f; Fin[lane][7] = 0.0f;   // zero-pad K 6 -> 8
    }
    __syncthreads();
    gemm_tile<2, 2>(&Fin[0][0], 8, B1, &T0[0][0], 32, lane);          // lin1
    __syncthreads();
    silu_tile(&T0[0][0], 32, lane);
    __syncthreads();
    layernorm_tile(&T0[0][0], 32, &T1[0][0], HID, g1, nullptr, lane); // ln1
    __syncthreads();
    gemm_tile<5, 2>(&T1[0][0], HID, B2, &T0[0][0], 32, lane);         // lin2
    __syncthreads();
    silu_tile(&T0[0][0], 32, lane);
    __syncthreads();

    long long g = row0 + row;
#pragma unroll
    for (int j = 0; j < 10; j++) xbuf[g * HID + h0 + j] = T0[row][h0 + j];
  }
}

// ===========================================================================
// K2: both LSTMs fused, step-synchronous.  One wave per 16 batch rows; the
// recurrence runs t = 0..127 with x updated in place (x -> x + h1 -> + h2).
// All four weight matrices live as B-fragments in VGPRs for the whole loop
// (100 v2f = 200 VGPRs; wave32 allows 1024). 100 v_wmma per timestep per wave.
// ===========================================================================
__global__ __launch_bounds__(32) void k_lstm2x(
    const float* __restrict__ ln2_g,
    const float* __restrict__ wih1, const float* __restrict__ whh1,
    const float* __restrict__ bih1, const float* __restrict__ bhh1,
    const float* __restrict__ ln3_g, const float* __restrict__ ln3_b,
    const float* __restrict__ wih2, const float* __restrict__ whh2,
    const float* __restrict__ bih2, const float* __restrict__ bhh2,
    float* __restrict__ xbuf) {
  __shared__ float Xs[16][20];
  __shared__ float As[16][20];
  __shared__ float H1[16][20];
  __shared__ float H2[16][20];
  __shared__ float Gs[16][80];
  int lane = threadIdx.x;
  int n0 = blockIdx.x * 16;

  // Stage all four weight matrices as WMMA B-fragments in registers.
  // All indices valid (Kw=20, OUT=80): unconditional 8B vector loads.
  v2f Fih1[5][5], Fhh1[5][5], Fih2[5][5], Fhh2[5][5];
  {
    int jj = lane & 15;
    int kb = (lane >> 4) << 1;
#pragma unroll
    for (int kc = 0; kc < 5; kc++)
#pragma unroll
      for (int ot = 0; ot < 5; ot++) {
        int off = (ot * 16 + jj) * HID + 4 * kc + kb;
        Fih1[kc][ot] = loadv2(wih1 + off);
        Fhh1[kc][ot] = loadv2(whh1 + off);
        Fih2[kc][ot] = loadv2(wih2 + off);
        Fhh2[kc][ot] = loadv2(whh2 + off);
      }
  }
  // Per-lane folded gate bias (bih + bhh), one per output tile (col = lane&15).
  float gb1[5], gb2[5];
#pragma unroll
  for (int ot = 0; ot < 5; ot++) {
    int j = ot * 16 + (lane & 15);
    gb1[ot] = bih1[j] + bhh1[j];
    gb2[ot] = bih2[j] + bhh2[j];
  }

  int row = lane >> 1, h0 = (lane & 1) * 10;
  float c1[10], c2[10];
#pragma unroll
  for (int j = 0; j < 10; j++) {
    c1[j] = 0.0f; c2[j] = 0.0f;
    H1[row][h0 + j] = 0.0f;
    H2[row][h0 + j] = 0.0f;
  }
  __syncthreads();

  for (int t = 0; t < LSEQ; t++) {
    float* xt = xbuf + ((long long)t * NBAT + n0) * HID;
#pragma unroll
    for (int j = 0; j < 10; j++) Xs[row][h0 + j] = xt[row * HID + h0 + j];
    if (t + 1 < LSEQ)   // hint next timestep's tile into cache (global_prefetch)
      __builtin_prefetch(xbuf + ((long long)(t + 1) * NBAT + n0) * HID + lane * 10, 0, 1);
    __syncthreads();

    layernorm_tile(&Xs[0][0], HID, &As[0][0], HID, ln2_g, nullptr, lane);
    __syncthreads();
    lstm_gates(&As[0][0], &H1[0][0], Fih1, Fhh1, gb1, &Gs[0][0], lane);
    __syncthreads();
    lstm_pointwise(&Gs[0][0], &H1[0][0], &Xs[0][0], c1, lane);   // x += h1
    __syncthreads();

    layernorm_tile(&Xs[0][0], HID, &As[0][0], HID, ln3_g, ln3_b, lane);
    __syncthreads();
    lstm_gates(&As[0][0], &H2[0][0], Fih2, Fhh2, gb2, &Gs[0][0], lane);
    __syncthreads();
    lstm_pointwise(&Gs[0][0], &H2[0][0], &Xs[0][0], c2, lane);   // x += h2

#pragma unroll
    for (int j = 0; j < 10; j++) xt[row * HID + h0 + j] = Xs[row][h0 + j];
    __syncthreads();
  }
}

// ===========================================================================
// K3: post-chain (ln4/lin3, ln5/lin4 residual, ln6/lin5) + the three heads.
// ===========================================================================
__global__ __launch_bounds__(32) void k_post(
    const float* __restrict__ xbuf,
    const float* __restrict__ g4,
    const float* __restrict__ w3, const float* __restrict__ b3,
    const float* __restrict__ g5,
    const float* __restrict__ w4, const float* __restrict__ b4,
    const float* __restrict__ g6,
    const float* __restrict__ w5, const float* __restrict__ b5,
    const float* __restrict__ wW, const float* __restrict__ wB,
    const float* __restrict__ sW, const float* __restrict__ sB,
    const float* __restrict__ dW, const float* __restrict__ dB,
    float* __restrict__ out) {
  __shared__ float Xs[16][20];
  __shared__ float T0[16][20];
  __shared__ float T1[16][32];   // GEMM dest: stride 32, unguarded stores
  int lane = threadIdx.x;
  int row = lane >> 1, h0 = (lane & 1) * 10;

  BW<5, 2> B3; load_bw(B3, w3, HID, HID, b3, lane);
  BW<5, 2> B4; load_bw(B4, w4, HID, HID, b4, lane);
  BW<5, 2> B5; load_bw(B5, w5, HID, HID, b5, lane);

  for (int tt = 0; tt < TPB; tt++) {
    long long row0 = ((long long)blockIdx.x * TPB + tt) * 16;
    __syncthreads();
#pragma unroll
    for (int j = 0; j < 10; j++) Xs[row][h0 + j] = xbuf[(row0 + row) * HID + h0 + j];
    __syncthreads();

    layernorm_tile(&Xs[0][0], HID, &T0[0][0], HID, g4, nullptr, lane);  // ln4
    __syncthreads();
    gemm_tile<5, 2>(&T0[0][0], HID, B3, &T1[0][0], 32, lane);           // lin3
    __syncthreads();
    silu_tile(&T1[0][0], 32, lane);
    __syncthreads();
    layernorm_tile(&T1[0][0], 32, &T0[0][0], HID, g5, nullptr, lane);   // ln5
    __syncthreads();
    gemm_tile<5, 2>(&T0[0][0], HID, B4, &T1[0][0], 32, lane);           // lin4
    __syncthreads();
    silu_tile(&T1[0][0], 32, lane);
    __syncthreads();
#pragma unroll
    for (int j = 0; j < 10; j++) Xs[row][h0 + j] += T1[row][h0 + j];    // x += h
    __syncthreads();
    layernorm_tile(&Xs[0][0], HID, &T0[0][0], HID, g6, nullptr, lane);  // ln6
    __syncthreads();
    gemm_tile<5, 2>(&T0[0][0], HID, B5, &T1[0][0], 32, lane);           // lin5
    __syncthreads();
    silu_tile(&T1[0][0], 32, lane);
    __syncthreads();

    // Heads (C=3: scalar dots). lanes 0-15 do w(softmax)+s for their row,
    // lanes 16-31 do d for the same rows.
    int r = lane & 15;
    long long g = row0 + r;
    float x[20];
#pragma unroll
    for (int j = 0; j < 20; j++) x[j] = T1[r][j];

    if (lane < 16) {
      float z[3];
#pragma unroll
      for (int c = 0; c < 3; c++) {
        float acc = wB[c];
#pragma unroll
        for (int j = 0; j < 20; j++) acc += x[j] * wW[c * 20 + j];
        z[c] = acc;
      }
      float m = fmaxf(z[0], fmaxf(z[1], z[2]));
      float e0 = __expf(z[0] - m), e1 = __expf(z[1] - m), e2 = __expf(z[2] - m);
      float inv = rcpf_(e0 + e1 + e2);
      out[g * 3 + 0] = e0 * inv;
      out[g * 3 + 1] = e1 * inv;
      out[g * 3 + 2] = e2 * inv;
#pragma unroll
      for (int c = 0; c < 3; c++) {
        float acc = sB[c];
#pragma unroll
        for (int j = 0; j < 20; j++) acc += x[j] * sW[c * 20 + j];
        acc = fminf(fmaxf(acc, -25.0f), 25.0f);
        out[LN3 + g * 3 + c] = __expf(acc);
      }
    } else {
#pragma unroll
      for (int c = 0; c < 3; c++) {
        float acc = dB[c];
#pragma unroll
        for (int j = 0; j < 20; j++) acc += x[j] * dW[c * 20 + j];
        acc = fminf(fmaxf(acc, -25.0f), 25.0f);
        out[2LL * LN3 + g * 3 + c] = __expf(acc);
      }
    }
  }
}

// ===========================================================================
extern "C" void kernel_launch(void* const* d_in, const int* in_sizes, int n_in,
                              void* d_out, int out_size, void* d_ws, size_t ws_size,
                              hipStream_t stream) {
  (void)in_sizes; (void)n_in; (void)out_size; (void)ws_size;
  const float* x_lni     = (const float*)d_in[0];
  const float* imean     = (const float*)d_in[1];
  const float* istd      = (const float*)d_in[2];
  const float* lin1_w    = (const float*)d_in[3];
  const float* lin1_b    = (const float*)d_in[4];
  const float* ln1_g     = (const float*)d_in[5];
  const float* lin2_w    = (const float*)d_in[6];
  const float* lin2_b    = (const float*)d_in[7];
  const float* ln2_g     = (const float*)d_in[8];
  const float* lstm1_wih = (const float*)d_in[9];
  const float* lstm1_whh = (const float*)d_in[10];
  const float* lstm1_bih = (const float*)d_in[11];
  const float* lstm1_bhh = (const float*)d_in[12];
  const float* ln3_g     = (const float*)d_in[13];
  const float* ln3_b     = (const float*)d_in[14];
  const float* lstm2_wih = (const float*)d_in[15];
  const float* lstm2_whh = (const float*)d_in[16];
  const float* lstm2_bih = (const float*)d_in[17];
  const float* lstm2_bhh = (const float*)d_in[18];
  const float* ln4_g     = (const float*)d_in[19];
  const float* lin3_w    = (const float*)d_in[20];
  const float* lin3_b    = (const float*)d_in[21];
  const float* ln5_g     = (const float*)d_in[22];
  const float* lin4_w    = (const float*)d_in[23];
  const float* lin4_b    = (const float*)d_in[24];
  const float* ln6_g     = (const float*)d_in[25];
  const float* lin5_w    = (const float*)d_in[26];
  const float* lin5_b    = (const float*)d_in[27];
  const float* w_fc_w    = (const float*)d_in[28];
  const float* w_fc_b    = (const float*)d_in[29];
  const float* s_fc_w    = (const float*)d_in[30];
  const float* s_fc_b    = (const float*)d_in[31];
  const float* d_fc_w    = (const float*)d_in[32];
  const float* d_fc_b    = (const float*)d_in[33];

  float* xbuf = (float*)d_ws;           // [L, N, 20] f32, reused in place
  float* out  = (float*)d_out;          // [3, L, N, 3] flat (w, s, d)

  dim3 blk(32);
  int blocks = (LSEQ * NBAT) / (16 * TPB);   // 8192 blocks, 8 tiles each

  k_pre<<<blocks, blk, 0, stream>>>(x_lni, imean, istd, lin1_w, lin1_b, ln1_g,
                                    lin2_w, lin2_b, xbuf);
  k_lstm2x<<<NBAT / 16, blk, 0, stream>>>(ln2_g, lstm1_wih, lstm1_whh, lstm1_bih,
                                          lstm1_bhh, ln3_g, ln3_b, lstm2_wih,
                                          lstm2_whh, lstm2_bih, lstm2_bhh, xbuf);
  k_post<<<blocks, blk, 0, stream>>>(xbuf, ln4_g, lin3_w, lin3_b, ln5_g, lin4_w,
                                     lin4_b, ln6_g, lin5_w, lin5_b, w_fc_w, w_fc_b,
                                     s_fc_w, s_fc_b, d_fc_w, d_fc_b, out);
}